// TransformerCritic_38027640438826
// MI455X (gfx1250) — compile-verified
//
#include <hip/hip_runtime.h>

// Problem constants (from reference): B=128, N=64, OBS=128, ACT=32, D=64
#define BB   128
#define NN   64
#define DD   64
#define OBSD 128
#define ACTD 32
#define OAD  160
#define BN   (BB*NN)        // 8192
#define BNN  (BB*NN*NN)     // 524288
#define LDT  66             // padded LDS tile stride: even (b64-aligned), 66%64=2 -> no bank conflicts

typedef __attribute__((ext_vector_type(2))) float v2f;
typedef __attribute__((ext_vector_type(8))) float v8f;

__device__ __forceinline__ float gelu_f(float x) {
  // exact erf-based GELU (torch nn.GELU default)
  return 0.5f * x * (1.0f + erff(x * 0.70710678118654752440f));
}

// One 16x16 fp32 output tile via chained V_WMMA_F32_16X16X4_F32.
// A-layout (16x4 f32): lanes 0-15 hold M=lane, K={0,1}; lanes 16-31 M=lane-16, K={2,3}.
// B-layout (4x16 f32): symmetric: lanes hold N=lane&15, K selected by half-wave.
template <typename FA, typename FB>
__device__ __forceinline__ v8f wmma_gemm16(int K, FA loadA, FB loadB) {
  const int lane = threadIdx.x & 31;
  const int klo  = (lane >> 4) << 1;   // 0 or 2
  const int idx  = lane & 15;
  v8f acc = {};
  for (int k = 0; k < K; k += 4) {
    v2f a, b;
    a.x = loadA(idx, k + klo);
    a.y = loadA(idx, k + klo + 1);
    b.x = loadB(k + klo,     idx);
    b.y = loadB(k + klo + 1, idx);
    acc = __builtin_amdgcn_wmma_f32_16x16x4_f32(
        /*neg_a=*/false, a, /*neg_b=*/false, b,
        /*c_mod=*/(short)0, acc, /*reuse_a=*/false, /*reuse_b=*/false);
  }
  return acc;
}

// C/D layout (f32 16x16): VGPR r: lanes 0-15 -> M=r,N=lane; lanes 16-31 -> M=r+8,N=lane-16.
#define TILE_STORE_BODY(EXPR_V, DSTSTMT)                                  \
  {                                                                       \
    const int lane_ = threadIdx.x & 31;                                   \
    const int half_ = lane_ >> 4;                                         \
    const int col_  = lane_ & 15;                                         \
    _Pragma("unroll")                                                     \
    for (int r_ = 0; r_ < 8; ++r_) {                                      \
      const int m = m0 + r_ + 8 * half_;                                  \
      const int n = n0 + col_;                                            \
      float v = (EXPR_V);                                                 \
      DSTSTMT;                                                            \
      (void)m; (void)n; (void)v;                                          \
    }                                                                     \
  }

// ---------------------------------------------------------------------------
// Kernel 1: per-batch attention front-end + cae/cae1.  grid = B, block = 256.
// ---------------------------------------------------------------------------
__global__ __launch_bounds__(256) void k1_attention(
    const float* __restrict__ states,
    const float* __restrict__ W_se, const float* __restrict__ b_se,
    const float* __restrict__ W_k,  const float* __restrict__ b_k,
    const float* __restrict__ W_q,  const float* __restrict__ b_q,
    const float* __restrict__ W_ca, const float* __restrict__ b_ca,
    const float* __restrict__ g2,   const float* __restrict__ beta2,
    const float* __restrict__ W_f1,
    float* __restrict__ out_weight,
    float* __restrict__ ws_weight,  float* __restrict__ ws_cae1) {
  __shared__ float s_se[NN * LDT];
  __shared__ float s_k [NN * LDT];
  __shared__ float s_q [NN * LDT];
  __shared__ float s_sc[NN * LDT];

  const int b    = blockIdx.x;
  const int wave = threadIdx.x >> 5;
  const float* stb = states + (size_t)b * NN * OBSD;

  // ---- se = gelu(states @ W_se + b_se) ----
  for (int rep = 0; rep < 2; ++rep) {
    const int t = wave + 8 * rep;
    const int m0 = (t >> 2) * 16, n0 = (t & 3) * 16;
    v8f acc = wmma_gemm16(OBSD,
        [&](int r, int k) { return stb[(m0 + r) * OBSD + k]; },
        [&](int k, int c) { return W_se[k * DD + (n0 + c)]; });
    TILE_STORE_BODY(gelu_f(acc[r_] + b_se[n]), s_se[m * LDT + n] = v);
  }
  __syncthreads();

  // ---- k_obs, q_obs ----
  for (int rep = 0; rep < 2; ++rep) {
    const int t = wave + 8 * rep;
    const int m0 = (t >> 2) * 16, n0 = (t & 3) * 16;
    v8f accK = wmma_gemm16(DD,
        [&](int r, int k) { return s_se[(m0 + r) * LDT + k]; },
        [&](int k, int c) { return W_k[k * DD + (n0 + c)]; });
    TILE_STORE_BODY(gelu_f(accK[r_] + b_k[n]), s_k[m * LDT + n] = v);
    v8f accQ = wmma_gemm16(DD,
        [&](int r, int k) { return s_se[(m0 + r) * LDT + k]; },
        [&](int k, int c) { return W_q[k * DD + (n0 + c)]; });
    TILE_STORE_BODY(gelu_f(accQ[r_] + b_q[n]), s_q[m * LDT + n] = v);
  }
  __syncthreads();

  // ---- scores = (q @ k^T) / 8 ----
  for (int rep = 0; rep < 2; ++rep) {
    const int t = wave + 8 * rep;
    const int m0 = (t >> 2) * 16, n0 = (t & 3) * 16;
    v8f acc = wmma_gemm16(DD,
        [&](int r, int k) { return s_q[(m0 + r) * LDT + k]; },
        [&](int k, int c) { return s_k[(n0 + c) * LDT + k]; });  // B = k_obs^T
    TILE_STORE_BODY(acc[r_] * 0.125f, s_sc[m * LDT + n] = v);
  }
  __syncthreads();

  // ---- softmax rows -> weight (4 threads per row, wave32 shuffles) ----
  {
    const int row = threadIdx.x >> 2;
    const int c   = threadIdx.x & 3;
    const int d0  = c * 16;
    float mx = -3.4e38f;
    #pragma unroll
    for (int j = 0; j < 16; ++j) mx = fmaxf(mx, s_sc[row * LDT + d0 + j]);
    mx = fmaxf(mx, __shfl_xor(mx, 1, 32));
    mx = fmaxf(mx, __shfl_xor(mx, 2, 32));
    float e[16];
    float sum = 0.f;
    #pragma unroll
    for (int j = 0; j < 16; ++j) { e[j] = __expf(s_sc[row * LDT + d0 + j] - mx); sum += e[j]; }
    sum += __shfl_xor(sum, 1, 32);
    sum += __shfl_xor(sum, 2, 32);
    const float inv = 1.0f / sum;
    #pragma unroll
    for (int j = 0; j < 16; ++j) {
      const float w = e[j] * inv;
      const size_t o = (size_t)(b * NN + row) * NN + d0 + j;
      ws_weight[o]  = w;
      out_weight[o] = w;
    }
  }

  // ---- cae_pre = gelu(states @ W_ca + b_ca) + se  (into s_q, which is free) ----
  for (int rep = 0; rep < 2; ++rep) {
    const int t = wave + 8 * rep;
    const int m0 = (t >> 2) * 16, n0 = (t & 3) * 16;
    v8f acc = wmma_gemm16(OBSD,
        [&](int r, int k) { return stb[(m0 + r) * OBSD + k]; },
        [&](int k, int c) { return W_ca[k * DD + (n0 + c)]; });
    TILE_STORE_BODY(gelu_f(acc[r_] + b_ca[n]) + s_se[m * LDT + n],
                    s_q[m * LDT + n] = v);
  }
  __syncthreads();

  // ---- LayerNorm(cae_pre; g2, beta2) in place (4 threads per row) ----
  {
    const int row = threadIdx.x >> 2;
    const int c   = threadIdx.x & 3;
    const int d0  = c * 16;
    float vals[16];
    float sum = 0.f;
    #pragma unroll
    for (int j = 0; j < 16; ++j) { vals[j] = s_q[row * LDT + d0 + j]; sum += vals[j]; }
    sum += __shfl_xor(sum, 1, 32);
    sum += __shfl_xor(sum, 2, 32);
    const float mean = sum * (1.0f / DD);
    float var = 0.f;
    #pragma unroll
    for (int j = 0; j < 16; ++j) { const float u = vals[j] - mean; var += u * u; }
    var += __shfl_xor(var, 1, 32);
    var += __shfl_xor(var, 2, 32);
    const float inv = rsqrtf(var * (1.0f / DD) + 1e-5f);
    #pragma unroll
    for (int j = 0; j < 16; ++j)
      s_q[row * LDT + d0 + j] = (vals[j] - mean) * inv * g2[d0 + j] + beta2[d0 + j];
  }
  __syncthreads();

  // ---- cae1 = cae @ W_f1[0:64,:]  -> ws ----
  for (int rep = 0; rep < 2; ++rep) {
    const int t = wave + 8 * rep;
    const int m0 = (t >> 2) * 16, n0 = (t & 3) * 16;
    v8f acc = wmma_gemm16(DD,
        [&](int r, int k) { return s_q[(m0 + r) * LDT + k]; },
        [&](int k, int c) { return W_f1[k * DD + (n0 + c)]; });
    TILE_STORE_BODY(acc[r_], ws_cae1[(size_t)(b * NN + m) * DD + n] = v);
  }
}

// ---------------------------------------------------------------------------
// Kernel 2: oape_act/pol, dav, base = weight @ av_act.  grid = B, block = 256.
// ---------------------------------------------------------------------------
__global__ __launch_bounds__(256) void k2_pairs(
    const float* __restrict__ states,
    const float* __restrict__ policies, const float* __restrict__ actions,
    const float* __restrict__ W_sap, const float* __restrict__ b_sap,
    const float* __restrict__ W_av,  const float* __restrict__ b_av,
    const float* __restrict__ ws_weight,
    float* __restrict__ ws_oact, float* __restrict__ ws_opol,
    float* __restrict__ ws_dav,  float* __restrict__ ws_base) {
  __shared__ float s_oape[NN * LDT];
  __shared__ float s_ava [NN * LDT];
  __shared__ float s_avp [NN * LDT];
  __shared__ float s_w   [NN * LDT];

  const int b    = blockIdx.x;
  const int wave = threadIdx.x >> 5;
  const float* stb  = states   + (size_t)b * NN * OBSD;
  const float* actb = actions  + (size_t)b * NN * ACTD;
  const float* polb = policies + (size_t)b * NN * ACTD;

  for (int pass = 0; pass < 2; ++pass) {
    const float* xb = (pass == 0) ? actb : polb;
    float* ws_oape  = (pass == 0) ? ws_oact : ws_opol;
    // oape = gelu(concat(states, act/pol) @ W_sap + b_sap)   (K = 160)
    for (int rep = 0; rep < 2; ++rep) {
      const int t = wave + 8 * rep;
      const int m0 = (t >> 2) * 16, n0 = (t & 3) * 16;
      v8f acc = wmma_gemm16(OAD,
          [&](int r, int k) {
            const int m = m0 + r;
            return (k < OBSD) ? stb[m * OBSD + k] : xb[m * ACTD + (k - OBSD)];
          },
          [&](int k, int c) { return W_sap[k * DD + (n0 + c)]; });
      TILE_STORE_BODY(gelu_f(acc[r_] + b_sap[n]),
                      { s_oape[m * LDT + n] = v;
                        ws_oape[(size_t)(b * NN + m) * DD + n] = v; });
    }
    __syncthreads();
    // av = gelu(oape @ W_av + b_av)
    float* s_av = (pass == 0) ? s_ava : s_avp;
    for (int rep = 0; rep < 2; ++rep) {
      const int t = wave + 8 * rep;
      const int m0 = (t >> 2) * 16, n0 = (t & 3) * 16;
      v8f acc = wmma_gemm16(DD,
          [&](int r, int k) { return s_oape[(m0 + r) * LDT + k]; },
          [&](int k, int c) { return W_av[k * DD + (n0 + c)]; });
      TILE_STORE_BODY(gelu_f(acc[r_] + b_av[n]), s_av[m * LDT + n] = v);
    }
    __syncthreads();
  }

  // dav = av_pol - av_act; stage weight[b] into LDS
  for (int i = threadIdx.x; i < NN * DD; i += 256) {
    const int row = i >> 6, d = i & 63;
    ws_dav[(size_t)b * NN * DD + i] = s_avp[row * LDT + d] - s_ava[row * LDT + d];
    s_w[row * LDT + d] = ws_weight[(size_t)b * NN * NN + i];
  }
  __syncthreads();

  // base = weight[b] @ av_act[b]
  for (int rep = 0; rep < 2; ++rep) {
    const int t = wave + 8 * rep;
    const int m0 = (t >> 2) * 16, n0 = (t & 3) * 16;
    v8f acc = wmma_gemm16(NN,
        [&](int r, int k) { return s_w[(m0 + r) * LDT + k]; },
        [&](int k, int c) { return s_ava[k * LDT + (n0 + c)]; });
    TILE_STORE_BODY(acc[r_], ws_base[(size_t)(b * NN + m) * DD + n] = v);
  }
}

// ---------------------------------------------------------------------------
// Kernel 3: fused nf-LN -> f1 -> f2 -> f3.  grid = B*N (one block per (b,x)).
// Weights staged transposed into LDS so B fetches are contiguous ds_load_b64.
// ---------------------------------------------------------------------------
__global__ __launch_bounds__(256) void k3_value(
    const float* __restrict__ ws_weight, const float* __restrict__ ws_base,
    const float* __restrict__ ws_cae1,
    const float* __restrict__ ws_oact,   const float* __restrict__ ws_opol,
    const float* __restrict__ ws_dav,
    const float* __restrict__ g1, const float* __restrict__ beta1,
    const float* __restrict__ W_f1, const float* __restrict__ b_f1,
    const float* __restrict__ W_f2, const float* __restrict__ b_f2,
    const float* __restrict__ W_f3, const float* __restrict__ b_f3,
    float* __restrict__ out_value) {
  __shared__ float s_nf[NN * LDT];
  __shared__ float s_h1[NN * LDT];
  __shared__ float s_h2[NN * LDT];
  __shared__ float s_w [NN * LDT];   // transposed weight staging (reused for W_f1b then W_f2)

  const int bx = blockIdx.x;       // b*64 + x
  const int b  = bx >> 6;
  const int x  = bx & 63;
  const int wave = threadIdx.x >> 5;
  const float* W1b = W_f1 + DD * DD;

  // stage W_f1b transposed: s_w[n*LDT+k] = W1b[k*64+n]  (coalesced global reads)
  for (int i = threadIdx.x; i < DD * DD; i += 256) {
    const int k = i >> 6, n = i & 63;
    s_w[n * LDT + k] = W1b[i];
  }

  // nf[y,:] = LN( oape_sel(y) + base[b,x] + weight[b,x,y]*dav[b,y] ; g1, beta1 )
  // 4 threads per row, wave32 shuffle reductions.
  {
    const int y  = threadIdx.x >> 2;
    const int c  = threadIdx.x & 3;
    const int d0 = c * 16;
    const float wxy = ws_weight[(size_t)bx * NN + y];
    const float* oape = ((y == x) ? ws_opol : ws_oact) + (size_t)(b * NN + y) * DD;
    const float* dv   = ws_dav  + (size_t)(b * NN + y) * DD;
    const float* bs   = ws_base + (size_t)bx * DD;
    float vals[16];
    float sum = 0.f;
    #pragma unroll
    for (int j = 0; j < 16; ++j) {
      const float v = oape[d0 + j] + bs[d0 + j] + wxy * dv[d0 + j];
      vals[j] = v;
      sum += v;
    }
    sum += __shfl_xor(sum, 1, 32);
    sum += __shfl_xor(sum, 2, 32);
    const float mean = sum * (1.0f / DD);
    float var = 0.f;
    #pragma unroll
    for (int j = 0; j < 16; ++j) { const float u = vals[j] - mean; var += u * u; }
    var += __shfl_xor(var, 1, 32);
    var += __shfl_xor(var, 2, 32);
    const float inv = rsqrtf(var * (1.0f / DD) + 1e-5f);
    #pragma unroll
    for (int j = 0; j < 16; ++j)
      s_nf[y * LDT + d0 + j] = (vals[j] - mean) * inv * g1[d0 + j] + beta1[d0 + j];
  }
  __syncthreads();

  // h1 = gelu(nf @ W_f1[64:128,:] + cae1[b,x] + b_f1)
  const float* c1 = ws_cae1 + (size_t)bx * DD;
  for (int rep = 0; rep < 2; ++rep) {
    const int t = wave + 8 * rep;
    const int m0 = (t >> 2) * 16, n0 = (t & 3) * 16;
    v8f acc = wmma_gemm16(DD,
        [&](int r, int k) { return s_nf[(m0 + r) * LDT + k]; },
        [&](int k, int c) { return s_w[(n0 + c) * LDT + k]; });   // transposed -> b64 pair
    TILE_STORE_BODY(gelu_f(acc[r_] + b_f1[n] + c1[n]), s_h1[m * LDT + n] = v);
  }
  __syncthreads();

  // restage W_f2 transposed into s_w (h1 GEMM done reading it)
  for (int i = threadIdx.x; i < DD * DD; i += 256) {
    const int k = i >> 6, n = i & 63;
    s_w[n * LDT + k] = W_f2[i];
  }
  __syncthreads();

  // h2 = gelu(h1 @ W_f2 + b_f2)
  for (int rep = 0; rep < 2; ++rep) {
    const int t = wave + 8 * rep;
    const int m0 = (t >> 2) * 16, n0 = (t & 3) * 16;
    v8f acc = wmma_gemm16(DD,
        [&](int r, int k) { return s_h1[(m0 + r) * LDT + k]; },
        [&](int k, int c) { return s_w[(n0 + c) * LDT + k]; });
    TILE_STORE_BODY(gelu_f(acc[r_] + b_f2[n]), s_h2[m * LDT + n] = v);
  }
  __syncthreads();

  // value[y] = h2[y] . W_f3 + b_f3   (4 threads per row)
  {
    const int y  = threadIdx.x >> 2;
    const int c  = threadIdx.x & 3;
    const int d0 = c * 16;
    float p = 0.f;
    #pragma unroll
    for (int j = 0; j < 16; ++j) p += s_h2[y * LDT + d0 + j] * W_f3[d0 + j];
    p += __shfl_xor(p, 1, 32);
    p += __shfl_xor(p, 2, 32);
    if (c == 0) out_value[(size_t)bx * NN + y] = p + b_f3[0];
  }
}

// ---------------------------------------------------------------------------
extern "C" void kernel_launch(void* const* d_in, const int* in_sizes, int n_in,
                              void* d_out, int out_size, void* d_ws, size_t ws_size,
                              hipStream_t stream) {
  const float* states   = (const float*)d_in[0];
  const float* policies = (const float*)d_in[1];
  const float* actions  = (const float*)d_in[2];
  const float* W_se = (const float*)d_in[3];  const float* b_se = (const float*)d_in[4];
  const float* W_k  = (const float*)d_in[5];  const float* b_k  = (const float*)d_in[6];
  const float* W_q  = (const float*)d_in[7];  const float* b_q  = (const float*)d_in[8];
  const float* W_sap= (const float*)d_in[9];  const float* b_sap= (const float*)d_in[10];
  const float* W_av = (const float*)d_in[11]; const float* b_av = (const float*)d_in[12];
  const float* g1   = (const float*)d_in[13]; const float* beta1= (const float*)d_in[14];
  const float* W_ca = (const float*)d_in[15]; const float* b_ca = (const float*)d_in[16];
  const float* g2   = (const float*)d_in[17]; const float* beta2= (const float*)d_in[18];
  const float* W_f1 = (const float*)d_in[19]; const float* b_f1 = (const float*)d_in[20];
  const float* W_f2 = (const float*)d_in[21]; const float* b_f2 = (const float*)d_in[22];
  const float* W_f3 = (const float*)d_in[23]; const float* b_f3 = (const float*)d_in[24];

  float* out_value  = (float*)d_out;           // [B,N,N,1] = 524288 floats
  float* out_weight = out_value + BNN;         // [B,N,N]   = 524288 floats

  float* ws        = (float*)d_ws;             // 3,145,728 floats = 12 MB
  float* ws_weight = ws;
  float* ws_cae1   = ws +  524288;
  float* ws_oact   = ws + 1048576;
  float* ws_opol   = ws + 1572864;
  float* ws_dav    = ws + 2097152;
  float* ws_base   = ws + 2621440;

  k1_attention<<<BB, 256, 0, stream>>>(states, W_se, b_se, W_k, b_k, W_q, b_q,
                                       W_ca, b_ca, g2, beta2, W_f1,
                                       out_weight, ws_weight, ws_cae1);
  k2_pairs<<<BB, 256, 0, stream>>>(states, policies, actions, W_sap, b_sap,
                                   W_av, b_av, ws_weight,
                                   ws_oact, ws_opol, ws_dav, ws_base);
  k3_value<<<BN, 256, 0, stream>>>(ws_weight, ws_base, ws_cae1, ws_oact, ws_opol,
                                   ws_dav, g1, beta1, W_f1, b_f1, W_f2, b_f2,
                                   W_f3, b_f3, out_value);
}